// Trainable_Difference_Layer_21320217657991
// MI455X (gfx1250) — compile-verified
//
#include <hip/hip_runtime.h>
#include <math.h>

typedef __attribute__((ext_vector_type(2))) float v2f;
typedef __attribute__((ext_vector_type(8))) float v8f;

#define N_EX 4096
#define KF   256
#define P2   96
#define ROWS_PER_BLOCK 64   // 4 waves x 16 examples

// ---------------------------------------------------------------------------
// Kernel 1: per-frequency logdet of (W @ Sigma1 + I), clip done in log space.
// ---------------------------------------------------------------------------
__global__ __launch_bounds__(128) void det_kernel(const float* __restrict__ w,
                                                  const float* __restrict__ s1,
                                                  float* __restrict__ logdet) {
  __shared__ float A[P2][P2 + 1];
  __shared__ int spiv;
  const int f = blockIdx.x;
  const int tid = threadIdx.x;
  const float* wf = w + (size_t)f * P2 * P2;
  const float* sf = s1 + (size_t)f * P2 * P2;

  for (int e = tid; e < P2 * P2; e += 128) {
    const int i = e / P2, j = e % P2;
    float acc = (i == j) ? 1.0f : 0.0f;
    for (int p = 0; p < P2; ++p) acc += wf[i * P2 + p] * sf[p * P2 + j];
    A[i][j] = acc;
  }
  __syncthreads();

  float sign = 1.0f, logabs = 0.0f;  // meaningful on tid 0 only
  for (int k = 0; k < P2; ++k) {
    if (tid == 0) {
      int pr = k;
      float best = fabsf(A[k][k]);
      for (int i = k + 1; i < P2; ++i) {
        const float v = fabsf(A[i][k]);
        if (v > best) { best = v; pr = i; }
      }
      spiv = pr;
    }
    __syncthreads();
    const int pr = spiv;
    if (pr != k) {
      for (int j = tid; j < P2; j += 128) {
        const float t = A[k][j]; A[k][j] = A[pr][j]; A[pr][j] = t;
      }
    }
    __syncthreads();
    const float piv = A[k][k];
    if (tid == 0) {
      if (pr != k)    sign = -sign;
      if (piv < 0.0f) sign = -sign;
      logabs += __logf(fabsf(piv));
    }
    const float inv = 1.0f / piv;
    for (int i = k + 1 + tid; i < P2; i += 128) {
      const float fct = A[i][k] * inv;
      for (int j = k; j < P2; ++j) A[i][j] -= fct * A[k][j];
    }
    __syncthreads();
  }
  if (tid == 0) {
    const float lmin = __logf(1e-5f), lmax = __logf(1e5f);
    logdet[f] = (sign < 0.0f) ? lmin : fminf(fmaxf(logabs, lmin), lmax);
  }
}

// ---------------------------------------------------------------------------
// Kernel 2: quad[n,f] = z^T W z via V_WMMA_F32_16X16X4_F32.
//  - W staged K-interleaved (Wq[k/4][q][k%4]) so a B fragment is one aligned
//    ds_load_b64 straight into an even VGPR pair (no packing movs, no bank
//    conflicts: half-waves hit banks {4m,4m+1} vs {4m+2,4m+3}).
//  - z tile staged with GLOBAL_LOAD_ASYNC_TO_LDS_B64 (ASYNCcnt path).
// ---------------------------------------------------------------------------
__global__ __launch_bounds__(128) void quad_kernel(const float* __restrict__ z,
                                                   const float* __restrict__ w,
                                                   float* __restrict__ quad) {
  __shared__ float Wq[P2 / 4][P2][4];            // 36,864 B, K-interleaved
  __shared__ float Zs[ROWS_PER_BLOCK][P2 + 2];   // 25,088 B, stride 98 (even)
  const int f  = blockIdx.y;
  const int n0 = blockIdx.x * ROWS_PER_BLOCK;
  const int tid = threadIdx.x;

  // --- Stage W_f, swizzling K into the fastest axis ---
  const float* wf = w + (size_t)f * P2 * P2;
  for (int e = tid * 4; e < P2 * P2; e += 128 * 4) {
    const float4 v = *(const float4*)(wf + e);
    const int k = e / P2, c = e % P2;   // row k (K dim), cols c..c+3 (q dim)
    const int t = k >> 2, s = k & 3;
    Wq[t][c + 0][s] = v.x;
    Wq[t][c + 1][s] = v.y;
    Wq[t][c + 2][s] = v.z;
    Wq[t][c + 3][s] = v.w;
  }

  // --- Stage 64 z rows via async global->LDS copies (8B granules) ---
  // Low 32 bits of a generic shared pointer are the LDS byte offset
  // (flat aperture rule).
  for (int u = tid; u < ROWS_PER_BLOCK * (P2 / 2); u += 128) {
    const int r  = u / (P2 / 2);
    const int c2 = u % (P2 / 2);
    const float* gp = z + ((size_t)(n0 + r) * KF + f) * P2 + c2 * 2;
    const unsigned lds = (unsigned)(size_t)&Zs[r][c2 * 2];
    asm volatile("global_load_async_to_lds_b64 %0, %1, off"
                 :: "v"(lds), "v"((unsigned long long)(size_t)gp)
                 : "memory");
  }
  asm volatile("s_wait_asynccnt 0" ::: "memory");
  __syncthreads();

  const int wave = tid >> 5;
  const int lane = tid & 31;
  const int m0   = wave * 16;        // this wave's example tile
  const int mloc = lane & 15;        // A: row M, B/C: col
  const int kh   = (lane >> 4) << 1; // half-wave K offset (0 or 2)

  v8f acc[6] = {};  // Y tile accumulators, q-tiles 0..5

  for (int k0 = 0; k0 < P2; k0 += 4) {
    const v2f a = *(const v2f*)&Zs[m0 + mloc][k0 + kh];  // ds_load_b64
    const float* wrow = &Wq[k0 >> 2][0][kh];
#pragma unroll
    for (int qt = 0; qt < 6; ++qt) {
      const v2f b = *(const v2f*)&wrow[(qt * 16 + mloc) * 4];  // ds_load_b64
      acc[qt] = __builtin_amdgcn_wmma_f32_16x16x4_f32(
          /*neg_a=*/false, a, /*neg_b=*/false, b,
          /*c_mod=*/(short)0, acc[qt], /*reuse_a=*/false, /*reuse_b=*/false);
    }
  }

  // --- Epilogue: quad[m] = sum_q Y[m,q] * z[m,q] ---
  // C/D layout: VGPR j holds row (j + 8*halfwave), column = lane%16.
  float part[8];
#pragma unroll
  for (int j = 0; j < 8; ++j) part[j] = 0.0f;
  const int mbase = m0 + ((lane >> 4) << 3);
#pragma unroll
  for (int qt = 0; qt < 6; ++qt) {
    const int q = qt * 16 + mloc;
#pragma unroll
    for (int j = 0; j < 8; ++j) part[j] += acc[qt][j] * Zs[mbase + j][q];
  }
#pragma unroll
  for (int off = 1; off < 16; off <<= 1) {
#pragma unroll
    for (int j = 0; j < 8; ++j) part[j] += __shfl_xor(part[j], off, 32);
  }
  if (mloc == 0) {
    const int rbase = n0 + mbase;
#pragma unroll
    for (int j = 0; j < 8; ++j) quad[(size_t)(rbase + j) * KF + f] = part[j];
  }
}

// ---------------------------------------------------------------------------
// Kernel 3: out[n] = sum_f (quad[n,f] - logdet[f]) + log(n1/n2). One wave / n.
// ---------------------------------------------------------------------------
__global__ __launch_bounds__(256) void final_kernel(const float* __restrict__ quad,
                                                    const float* __restrict__ logdet,
                                                    const int* __restrict__ n1,
                                                    const int* __restrict__ n2,
                                                    float* __restrict__ out) {
  const int gw   = (blockIdx.x * 256 + threadIdx.x) >> 5;
  const int lane = threadIdx.x & 31;
  if (gw >= N_EX) return;
  float s = 0.0f;
  for (int fb = lane; fb < KF; fb += 32)
    s += quad[(size_t)gw * KF + fb] - logdet[fb];
#pragma unroll
  for (int off = 16; off > 0; off >>= 1) s += __shfl_xor(s, off, 32);
  if (lane == 0) out[gw] = s + __logf((float)n1[0] / (float)n2[0]);
}

// ---------------------------------------------------------------------------
extern "C" void kernel_launch(void* const* d_in, const int* in_sizes, int n_in,
                              void* d_out, int out_size, void* d_ws, size_t ws_size,
                              hipStream_t stream) {
  const float* z  = (const float*)d_in[0];
  const float* s1 = (const float*)d_in[1];
  // d_in[2] (sigma_2) is unused by the reference computation.
  const float* w  = (const float*)d_in[3];
  const int*   n1 = (const int*)d_in[4];
  const int*   n2 = (const int*)d_in[5];
  float* out = (float*)d_out;

  // workspace: quad (N*K floats = 4 MB) then logdet (K floats)
  float* quad   = (float*)d_ws;
  float* logdet = quad + (size_t)N_EX * KF;

  det_kernel<<<KF, 128, 0, stream>>>(w, s1, logdet);
  quad_kernel<<<dim3(N_EX / ROWS_PER_BLOCK, KF), 128, 0, stream>>>(z, w, quad);
  final_kernel<<<(N_EX * 32) / 256, 256, 0, stream>>>(quad, logdet, n1, n2, out);
}